// SpatialFeatureExtractor_28741921145060
// MI455X (gfx1250) — compile-verified
//
#include <hip/hip_runtime.h>
#include <hip/hip_bf16.h>

typedef __attribute__((ext_vector_type(2))) float v2f;
typedef __attribute__((ext_vector_type(8))) float v8f;
typedef __attribute__((ext_vector_type(4))) int   v4i_t;
typedef __attribute__((address_space(1))) v4i_t   gv4i;   // global 16B vector
typedef __attribute__((address_space(3))) v4i_t   lv4i;   // LDS 16B vector

#define NC   64      // channels C
#define NT   500     // time T
#define TC   100     // T chunk size (5 chunks)
#define NCHK 5
#define CST  108     // chunk LDS stride in floats (16B-aligned rows, conflict-free frag loads)
#define ST   68      // 64x64 matrix LDS stride in floats (bank offset 4*row, conflict-free)
#define NBF  2560    // B*F = 64*40

#if defined(__has_builtin)
#if __has_builtin(__builtin_amdgcn_global_load_async_to_lds_b128)
#define HAVE_ASYNC_LDS 1
#endif
#endif
#ifndef HAVE_ASYNC_LDS
#define HAVE_ASYNC_LDS 0
#endif

// ---------------------------------------------------------------------------
// Copy one 64x100 chunk (6400 floats) of X into an LDS buffer.
// Async path: 1536 float4 via global_load_async_to_lds_b128 (exactly 6
// full-EXEC instructions per wave -> ASYNCcnt bookkeeping is uniform), plus a
// 64-float4 synchronous tail. Fallback: plain sync copy.
// ---------------------------------------------------------------------------
__device__ __forceinline__ void issue_chunk(const float* __restrict__ xb, int ch,
                                            float* __restrict__ dst, int tid)
{
#if HAVE_ASYNC_LDS
    #pragma unroll
    for (int k = 0; k < 6; ++k) {
        const int idx = tid + k * 256;            // uniform: all lanes active
        const int row = idx / 25, c4 = idx % 25;  // 25 float4 per row
        const float* g = xb + row * NT + ch * TC + c4 * 4;
        float* l = dst + row * CST + c4 * 4;
        __builtin_amdgcn_global_load_async_to_lds_b128(
            (gv4i*)g, (lv4i*)l, 0, 0);
    }
    if (tid < 64) {                               // tail: 64 float4, sync
        const int idx = 1536 + tid;
        const int row = idx / 25, c4 = idx % 25;
        *(float4*)(dst + row * CST + c4 * 4) =
            *(const float4*)(xb + row * NT + ch * TC + c4 * 4);
    }
#else
    for (int idx = tid; idx < NC * (TC / 4); idx += 256) {
        const int row = idx / 25, c4 = idx % 25;
        *(float4*)(dst + row * CST + c4 * 4) =
            *(const float4*)(xb + row * NT + ch * TC + c4 * 4);
    }
#endif
}

// ---------------------------------------------------------------------------
// Kernel 1: per (b,f) covariance via f32 WMMA Gram matrix + fused mean removal
//   cov = (X*X^T - (sum)(sum)^T / T) / (T-1)
// Double-buffered async chunk pipeline; one 256-thread block per (b,f).
// ---------------------------------------------------------------------------
__global__ __launch_bounds__(256) void cov_kernel(const float* __restrict__ x,
                                                  float* __restrict__ cov)
{
    __shared__ float cbuf[2][NC * CST];   // 2 x 27KB chunk buffers
    __shared__ float rsP[256];
    __shared__ float rowsum[NC];

    const int tid  = threadIdx.x;
    const int lane = tid & 31, wave = tid >> 5;
    const int half = lane >> 4, l15 = lane & 15;
    const int bf   = blockIdx.x;
    const float* xb = x + (size_t)bf * (NC * NT);

    // wave handles tiles t=wave -> (i0,j0) and t=wave+8 -> (i0+2,j0): shared B frag
    const int i0 = wave >> 2, j0 = wave & 3;

    v8f acc0 = {};
    v8f acc1 = {};
    float racc = 0.f;
    const int rrow = tid >> 2, rpart = tid & 3;

    issue_chunk(xb, 0, cbuf[0], tid);
    for (int ch = 0; ch < NCHK; ++ch) {
        const float* cur = cbuf[ch & 1];
        if (ch + 1 < NCHK) {
            issue_chunk(xb, ch + 1, cbuf[(ch + 1) & 1], tid);
#if HAVE_ASYNC_LDS
            asm volatile("s_wait_asynccnt 6" ::: "memory");  // chunk ch landed
#endif
        } else {
#if HAVE_ASYNC_LDS
            asm volatile("s_wait_asynccnt 0" ::: "memory");
#endif
        }
        __syncthreads();

        // row-sum partials (4 threads per row)
        {
            const float* r = &cur[rrow * CST + rpart * (TC / 4)];
            float s = 0.f;
            #pragma unroll
            for (int q = 0; q < TC / 4; ++q) s += r[q];
            racc += s;
        }
        // Gram accumulation: v_wmma_f32_16x16x4_f32 over 25 K-steps
        const float* A0 = &cur[(16 * i0 + l15) * CST + 2 * half];
        const float* A1 = &cur[(16 * (i0 + 2) + l15) * CST + 2 * half];
        const float* Bf = &cur[(16 * j0 + l15) * CST + 2 * half];
        #pragma unroll 5
        for (int k0 = 0; k0 < TC; k0 += 4) {
            v2f a0 = *(const v2f*)(A0 + k0);
            v2f a1 = *(const v2f*)(A1 + k0);
            v2f b  = *(const v2f*)(Bf + k0);
            acc0 = __builtin_amdgcn_wmma_f32_16x16x4_f32(false, a0, false, b,
                                                         (short)0, acc0, false, false);
            acc1 = __builtin_amdgcn_wmma_f32_16x16x4_f32(false, a1, false, b,
                                                         (short)0, acc1, false, false);
        }
        __syncthreads();   // all consumers done before buffer is re-filled
    }

    // reduce row sums
    rsP[tid] = racc;
    __syncthreads();
    if (tid < NC)
        rowsum[tid] = rsP[4 * tid] + rsP[4 * tid + 1] + rsP[4 * tid + 2] + rsP[4 * tid + 3];
    __syncthreads();

    float* cg = cov + (size_t)bf * (NC * NC);
    const float invT   = 1.f / (float)NT;
    const float invTm1 = 1.f / (float)(NT - 1);
    #pragma unroll
    for (int v = 0; v < 8; ++v) {
        const int m = v + 8 * half;
        {
            int row = 16 * i0 + m, col = 16 * j0 + l15;
            float g = acc0[v] - rowsum[row] * rowsum[col] * invT;
            cg[row * NC + col] = g * invTm1;
        }
        {
            int row = 16 * (i0 + 2) + m, col = 16 * j0 + l15;
            float g = acc1[v] - rowsum[row] * rowsum[col] * invT;
            cg[row * NC + col] = g * invTm1;
        }
    }
}

// ---------------------------------------------------------------------------
// 64x64 matmul in LDS, all waves of the block: D = scaleD*(A x B) + diagAdd*I
// Caller provides barriers before/after. D must differ from A and B.
// ---------------------------------------------------------------------------
__device__ __forceinline__ void mm64(const float* __restrict__ A,
                                     const float* __restrict__ B,
                                     float* __restrict__ D,
                                     float scaleD, float diagAdd)
{
    const int tid  = threadIdx.x;
    const int lane = tid & 31, wave = tid >> 5;
    const int half = lane >> 4, l15 = lane & 15;
    #pragma unroll
    for (int tt = 0; tt < 2; ++tt) {
        const int t = wave + 8 * tt;
        const int i = t >> 2, j = t & 3;
        v8f acc = {};
        const float* arow = A + (16 * i + l15) * ST + 2 * half;
        const float* bcol = B + 16 * j + l15;
        #pragma unroll
        for (int k0 = 0; k0 < 64; k0 += 4) {
            v2f a = *(const v2f*)(arow + k0);
            v2f b;
            b.x = bcol[(k0 + 2 * half) * ST];
            b.y = bcol[(k0 + 2 * half + 1) * ST];
            acc = __builtin_amdgcn_wmma_f32_16x16x4_f32(false, a, false, b,
                                                        (short)0, acc, false, false);
        }
        #pragma unroll
        for (int v = 0; v < 8; ++v) {
            const int row = 16 * i + v + 8 * half;
            const int col = 16 * j + l15;
            float val = acc[v] * scaleD;
            if (row == col) val += diagAdd;
            D[row * ST + col] = val;
        }
    }
}

// ---------------------------------------------------------------------------
// Kernel 2: per (b,f) SPD matrix log via inverse scaling-and-squaring:
//   A' = cov / s, s = trace/32                (spectrum ~ (0.2, 0.93))
//   4x Newton-Schulz matrix square roots      (10 coupled iters, 3 matmuls each)
//   log(M) by 8-term Mercator/Horner series in E = M - I
//   logm(cov) = 16*log(M) + ln(s)*I ; emit upper triangle.
// ---------------------------------------------------------------------------
__global__ __launch_bounds__(256) void logm_kernel(const float* __restrict__ cov,
                                                   float* __restrict__ out)
{
    __shared__ float bufs[4][NC * ST];   // ~68KB: fine within 320KB/WGP LDS
    __shared__ float red[2];             // [0]=1/s, [1]=ln(s)

    const int tid = threadIdx.x;
    const int bf  = blockIdx.x;
    const float* cg = cov + (size_t)bf * (NC * NC);

    float* Y = bufs[0];
    float* Z = bufs[1];
    float* W = bufs[2];
    float* U = bufs[3];

    for (int idx = tid; idx < NC * NC; idx += 256) {
        int r = idx >> 6, c = idx & 63;
        Y[r * ST + c] = cg[idx];
    }
    __syncthreads();
    if (tid == 0) {
        float tr = 0.f;
        for (int i = 0; i < NC; ++i) tr += Y[i * ST + i];
        float s = fmaxf(tr * (1.f / 32.f), 1e-30f);
        red[0] = 1.f / s;
        red[1] = logf(s);
    }
    __syncthreads();
    const float invs = red[0];
    for (int idx = tid; idx < NC * NC; idx += 256) {
        int r = idx >> 6, c = idx & 63;
        Y[r * ST + c] *= invs;
    }

    // 4 Newton-Schulz square-root stages
    for (int stage = 0; stage < 4; ++stage) {
        __syncthreads();
        for (int idx = tid; idx < NC * NC; idx += 256) {
            int r = idx >> 6, c = idx & 63;
            Z[r * ST + c] = (r == c) ? 1.f : 0.f;
        }
        __syncthreads();
        for (int it = 0; it < 10; ++it) {
            mm64(Z, Y, W, -0.5f, 1.5f);   // W = 1.5I - 0.5*Z*Y
            __syncthreads();
            mm64(Y, W, U, 1.f, 0.f);      // U = Y*W        (new Y)
            __syncthreads();
            mm64(W, Z, Y, 1.f, 0.f);      // oldY-buf = W*Z (new Z)
            __syncthreads();
            float* oy = Y; float* oz = Z;
            Y = U; Z = oy; U = oz;        // rotate buffers
        }
        // Y now holds sqrt of the stage input
    }

    __syncthreads();
    // E = M - I (in place, diagonal only changes)
    if (tid < NC) Y[tid * ST + tid] -= 1.f;
    __syncthreads();
    // Horner: H = c8*E + c7*I  (c_n = (-1)^(n+1)/n)
    for (int idx = tid; idx < NC * NC; idx += 256) {
        int r = idx >> 6, c = idx & 63;
        Z[r * ST + c] = -0.125f * Y[r * ST + c] + ((r == c) ? (1.f / 7.f) : 0.f);
    }
    __syncthreads();
    float* H = Z; float* fA = W; float* fB = U;
    for (int n = 6; n >= 1; --n) {
        const float cn = ((n & 1) ? 1.f : -1.f) / (float)n;
        mm64(Y, H, fA, 1.f, cn);          // H' = E*H + c_n*I
        __syncthreads();
        float* t = H; H = fA; fA = fB; fB = t;
    }
    mm64(Y, H, fA, 1.f, 0.f);             // L = E*H = log(M)
    __syncthreads();

    const float* L = fA;
    const float logs = red[1];
    float* ob = out + (size_t)bf * 2080;
    for (int p = tid; p < 2080; p += 256) {
        // invert p -> (i,j), i<=j, off(i) = i*(129-i)/2
        int i = (int)((129.0f - sqrtf(129.0f * 129.0f - 8.0f * (float)p)) * 0.5f);
        if (i < 0) i = 0;
        if (i > 63) i = 63;
        while (i > 0 && (i * (129 - i)) / 2 > p) --i;
        while (i < 63 && ((i + 1) * (129 - (i + 1))) / 2 <= p) ++i;
        const int j = i + (p - (i * (129 - i)) / 2);
        float v = 16.f * L[i * ST + j];
        if (i == j) v += logs;
        ob[p] = v;
    }
}

// ---------------------------------------------------------------------------
extern "C" void kernel_launch(void* const* d_in, const int* in_sizes, int n_in,
                              void* d_out, int out_size, void* d_ws, size_t ws_size,
                              hipStream_t stream)
{
    const float* x = (const float*)d_in[0];   // (64,40,64,500) f32
    float* out = (float*)d_out;               // (64,40,2080)  f32
    float* cov = (float*)d_ws;                // 2560*4096 floats (~42MB scratch)

    cov_kernel<<<dim3(NBF), dim3(256), 0, stream>>>(x, cov);
    logm_kernel<<<dim3(NBF), dim3(256), 0, stream>>>(cov, out);
}